// DepthNet_5626407158175
// MI455X (gfx1250) — compile-verified
//
#include <hip/hip_runtime.h>

#define Cn 32
#define Dn 64
#define Hn 128
#define Wn 160
#define HWn (Hn*Wn)           // 20480
#define CHWn (Cn*HWn)         // 655360
#define NVOX (Dn*HWn)         // 1310720

typedef __attribute__((ext_vector_type(2))) float v2f;
typedef __attribute__((ext_vector_type(4))) float v4f;
typedef __attribute__((ext_vector_type(8))) float v8f;

// ---------------------------------------------------------------------------
// Projection setup: fuse K@E for each view, invert ref, proj = src @ inv(ref).
// Writes per src view: rot[9], trans[3]  (12 floats) at rt + (v-1)*12.
// ---------------------------------------------------------------------------
__global__ void k_setup(const float* __restrict__ projm, float* __restrict__ rt) {
  if (threadIdx.x != 0 || blockIdx.x != 0) return;
  float fused[5][16];
  for (int v = 0; v < 5; ++v) {
    const float* E = projm + v * 32;       // proj_matrices[0][v][0]
    const float* K = projm + v * 32 + 16;  // proj_matrices[0][v][1]
    float out[16];
    for (int i = 0; i < 16; ++i) out[i] = E[i];
    for (int i = 0; i < 3; ++i)
      for (int j = 0; j < 4; ++j) {
        float s = 0.f;
        for (int k = 0; k < 3; ++k) s += K[i*4+k] * E[k*4+j];
        out[i*4+j] = s;
      }
    for (int i = 0; i < 16; ++i) fused[v][i] = out[i];
  }
  // Gauss-Jordan inverse of fused[0]
  float a[4][8];
  for (int i = 0; i < 4; ++i)
    for (int j = 0; j < 4; ++j) { a[i][j] = fused[0][i*4+j]; a[i][j+4] = (i==j) ? 1.f : 0.f; }
  for (int col = 0; col < 4; ++col) {
    int piv = col; float best = fabsf(a[col][col]);
    for (int r = col+1; r < 4; ++r) { float v = fabsf(a[r][col]); if (v > best) { best = v; piv = r; } }
    if (piv != col)
      for (int j = 0; j < 8; ++j) { float t = a[col][j]; a[col][j] = a[piv][j]; a[piv][j] = t; }
    float inv = 1.f / a[col][col];
    for (int j = 0; j < 8; ++j) a[col][j] *= inv;
    for (int r = 0; r < 4; ++r) if (r != col) {
      float f = a[r][col];
      for (int j = 0; j < 8; ++j) a[r][j] -= f * a[col][j];
    }
  }
  float invref[16];
  for (int i = 0; i < 4; ++i) for (int j = 0; j < 4; ++j) invref[i*4+j] = a[i][j+4];
  for (int sv = 1; sv < 5; ++sv) {
    float P[16];
    for (int i = 0; i < 4; ++i)
      for (int j = 0; j < 4; ++j) {
        float s = 0.f;
        for (int k = 0; k < 4; ++k) s += fused[sv][i*4+k] * invref[k*4+j];
        P[i*4+j] = s;
      }
    float* o = rt + (sv-1)*12;
    o[0]=P[0]; o[1]=P[1]; o[2]=P[2];
    o[3]=P[4]; o[4]=P[5]; o[5]=P[6];
    o[6]=P[8]; o[7]=P[9]; o[8]=P[10];
    o[9]=P[3]; o[10]=P[7]; o[11]=P[11];
  }
}

// ---------------------------------------------------------------------------
// (C,H,W) -> (H*W, C) transpose so bilinear taps become contiguous 128B loads.
// ---------------------------------------------------------------------------
__global__ void k_transpose(const float* __restrict__ fea, float* __restrict__ feaT) {
  int idx = blockIdx.x * blockDim.x + threadIdx.x;   // over C*HW
  int c = idx / HWn;
  int p = idx - c * HWn;
  feaT[(size_t)p * Cn + c] = fea[idx];
}

__global__ void k_init(float* __restrict__ simSum, float* __restrict__ wSum) {
  int idx = blockIdx.x * blockDim.x + threadIdx.x;
  simSum[idx] = 0.f;
  if (idx < HWn) wSum[idx] = 1e-5f;
}

// ---------------------------------------------------------------------------
// Homography warp + bilinear sample + 32-channel similarity dot. 1 thread/voxel.
// ---------------------------------------------------------------------------
__global__ void k_warp_sim(const float* __restrict__ srcT, const float* __restrict__ refT,
                           const float* __restrict__ depths, const float* __restrict__ rt,
                           float* __restrict__ simTmp) {
  int idx = blockIdx.x * blockDim.x + threadIdx.x;   // voxel = d*HW + p
  int d = idx / HWn;
  int p = idx - d * HWn;
  int y = p / Wn;
  int x = p - y * Wn;
  float r0=rt[0], r1=rt[1], r2=rt[2], r3=rt[3], r4=rt[4], r5=rt[5], r6=rt[6], r7=rt[7], r8=rt[8];
  float tx=rt[9], ty=rt[10], tz=rt[11];
  float fx = (float)x, fy = (float)y;
  float X0 = r0*fx + r1*fy + r2;
  float Y0 = r3*fx + r4*fy + r5;
  float Z0 = r6*fx + r7*fy + r8;
  float dep = depths[idx];
  float pw = Z0*dep + tz;
  float px = (X0*dep + tx) / pw;
  float py = (Y0*dep + ty) / pw;
  float x0 = floorf(px), y0 = floorf(py);
  float wx1 = px - x0, wy1 = py - y0;
  float wx0 = 1.f - wx1, wy0 = 1.f - wy1;

  v4f rvec[8];
  const v4f* rp = (const v4f*)(refT + (size_t)p * Cn);
  #pragma unroll
  for (int i = 0; i < 8; ++i) rvec[i] = rp[i];

  float acc = 0.f;
  auto tap = [&](float xi, float yi, float wgt) {
    bool inb = (xi >= 0.f) && (xi <= (float)(Wn-1)) && (yi >= 0.f) && (yi <= (float)(Hn-1));
    if (!inb) return;
    int xc = (int)xi, yc = (int)yi;
    const v4f* sp = (const v4f*)(srcT + (size_t)(yc*Wn + xc) * Cn);
    float dot = 0.f;
    #pragma unroll
    for (int i = 0; i < 8; ++i) {
      v4f s = sp[i];
      dot += s.x*rvec[i].x + s.y*rvec[i].y + s.z*rvec[i].z + s.w*rvec[i].w;
    }
    acc += wgt * dot;
  };
  tap(x0,     y0,     wx0*wy0);
  tap(x0+1.f, y0,     wx1*wy0);
  tap(x0,     y0+1.f, wx0*wy1);
  tap(x0+1.f, y0+1.f, wx1*wy1);
  simTmp[idx] = acc * (1.f/32.f);
}

// ---------------------------------------------------------------------------
// Pixelwise net via WMMA f32 16x16x4. Each wave handles 16 voxels per step:
//   layer0: h0[j] = relu(sim * w0[j]*g0[j]/sqrt(1+eps) + b0[j])   (j=0..15)
//   layer1: Y(8x16vox) = W1(8x16) @ H0(16x16vox)  -> 4 chained K=4 WMMAs
//   layer2 + sigmoid on lanes 0..15.
// A layout: lanes 0-15 carry M=0..15 {K=c*4+0, c*4+1}; lanes 16-31 {K+2, K+3}.
// B layout: lanes 0-15 carry N with {K=c*4+0, +1};     lanes 16-31 {K+2, K+3}.
// ---------------------------------------------------------------------------
__global__ void k_mlp(const float* __restrict__ simTmp,
                      const float* __restrict__ w0, const float* __restrict__ g0,
                      const float* __restrict__ b0, const float* __restrict__ w1,
                      const float* __restrict__ g1, const float* __restrict__ b1,
                      const float* __restrict__ w2, const float* __restrict__ bias2,
                      float* __restrict__ sigTmp) {
  const float rinv = 1.0f / sqrtf(1.0f + 1e-5f);   // BN eval scale
  int lane = threadIdx.x & 31;
  int waveGlobal = blockIdx.x * (blockDim.x >> 5) + (threadIdx.x >> 5);
  int m = lane & 15;
  int koff = (lane < 16) ? 0 : 2;

  float s0a[4], s0b[4], bb0a[4], bb0b[4], a0[4], a1[4];
  #pragma unroll
  for (int c = 0; c < 4; ++c) {
    int j0 = c*4 + koff;
    s0a[c]  = w0[j0]   * g0[j0]   * rinv;  bb0a[c] = b0[j0];
    s0b[c]  = w0[j0+1] * g0[j0+1] * rinv;  bb0b[c] = b0[j0+1];
    a0[c] = (m < 8) ? w1[m*16 + j0]     : 0.f;
    a1[c] = (m < 8) ? w1[m*16 + j0 + 1] : 0.f;
  }
  float s1k[8], b1k[8], w2k[8];
  #pragma unroll
  for (int k = 0; k < 8; ++k) { s1k[k] = g1[k]*rinv; b1k[k] = b1[k]; w2k[k] = w2[k]; }
  float bias = bias2[0];

  const int GROUPS = 8;                       // 128 voxels per wave
  size_t base = (size_t)waveGlobal * (16 * GROUPS);
  for (int g = 0; g < GROUPS; ++g) {
    size_t vox = base + (size_t)g*16 + m;
    float sim = simTmp[vox];
    v8f acc = {};
    #pragma unroll
    for (int c = 0; c < 4; ++c) {
      float h0 = fmaxf(sim * s0a[c] + bb0a[c], 0.f);
      float h1 = fmaxf(sim * s0b[c] + bb0b[c], 0.f);
      v2f A = {a0[c], a1[c]};
      v2f Bm = {h0, h1};
      acc = __builtin_amdgcn_wmma_f32_16x16x4_f32(false, A, false, Bm,
                                                  (short)0, acc, false, false);
    }
    if (lane < 16) {   // lanes 0-15 hold rows M=0..7 in acc[0..7]
      float out = bias;
      #pragma unroll
      for (int k = 0; k < 8; ++k)
        out += w2k[k] * fmaxf(acc[k]*s1k[k] + b1k[k], 0.f);
      sigTmp[vox] = 1.f / (1.f + expf(-out));
    }
  }
}

// ---------------------------------------------------------------------------
// Per pixel: vw = max_d sigmoid;  simSum += sim*vw;  wSum += vw.
// ---------------------------------------------------------------------------
__global__ void k_vw_accum(const float* __restrict__ simTmp, const float* __restrict__ sigTmp,
                           float* __restrict__ simSum, float* __restrict__ wSum,
                           float* __restrict__ vwOut) {
  int p = blockIdx.x * blockDim.x + threadIdx.x;
  if (p >= HWn) return;
  float vw = 0.f;   // sigmoid > 0
  for (int d = 0; d < Dn; ++d) vw = fmaxf(vw, sigTmp[d*HWn + p]);
  vwOut[p] = vw;
  for (int d = 0; d < Dn; ++d) simSum[d*HWn + p] += simTmp[d*HWn + p] * vw;
  wSum[p] += vw;
}

// ---------------------------------------------------------------------------
// Final: similarity = simSum/wSum; WTA depth of similarity; cost-reg MLP,
// softmax over D -> prob_volume, WTA depth, confidence = max prob = 1/sumexp.
// ---------------------------------------------------------------------------
__global__ void k_final(const float* __restrict__ simSum, const float* __restrict__ wSum,
                        const float* __restrict__ depths,
                        const float* __restrict__ cw0, const float* __restrict__ cb0,
                        const float* __restrict__ cw1, const float* __restrict__ cb1,
                        float* __restrict__ depthOut, float* __restrict__ simDepthOut,
                        float* __restrict__ confOut, float* __restrict__ probOut) {
  int p = blockIdx.x * blockDim.x + threadIdx.x;
  if (p >= HWn) return;
  float winv = 1.f / wSum[p];
  float w0l[8], b0l[8], w1l[8];
  #pragma unroll
  for (int k = 0; k < 8; ++k) { w0l[k] = cw0[k]; b0l[k] = cb0[k]; w1l[k] = cw1[k]; }
  float cb = cb1[0];

  float smax = -INFINITY, crmax = -INFINITY;
  int sarg = 0, crarg = 0;
  for (int d = 0; d < Dn; ++d) {
    float s = simSum[d*HWn + p] * winv;
    float cr = cb;
    #pragma unroll
    for (int k = 0; k < 8; ++k) cr += w1l[k] * fmaxf(w0l[k]*s + b0l[k], 0.f);
    if (s  > smax)  { smax  = s;  sarg  = d; }
    if (cr > crmax) { crmax = cr; crarg = d; }
  }
  float sumexp = 0.f;
  for (int d = 0; d < Dn; ++d) {
    float s = simSum[d*HWn + p] * winv;
    float cr = cb;
    #pragma unroll
    for (int k = 0; k < 8; ++k) cr += w1l[k] * fmaxf(w0l[k]*s + b0l[k], 0.f);
    sumexp += expf(cr - crmax);
  }
  float rs = 1.f / sumexp;
  for (int d = 0; d < Dn; ++d) {
    float s = simSum[d*HWn + p] * winv;
    float cr = cb;
    #pragma unroll
    for (int k = 0; k < 8; ++k) cr += w1l[k] * fmaxf(w0l[k]*s + b0l[k], 0.f);
    probOut[d*HWn + p] = expf(cr - crmax) * rs;
  }
  depthOut[p]    = depths[crarg*HWn + p];
  simDepthOut[p] = depths[sarg*HWn + p];
  confOut[p]     = rs;
}

// ---------------------------------------------------------------------------
extern "C" void kernel_launch(void* const* d_in, const int* in_sizes, int n_in,
                              void* d_out, int out_size, void* d_ws, size_t ws_size,
                              hipStream_t stream) {
  (void)in_sizes; (void)n_in; (void)out_size; (void)ws_size;
  const float* features = (const float*)d_in[0];   // (5,1,32,128,160)
  const float* projm    = (const float*)d_in[1];   // (1,5,2,4,4)
  const float* depths   = (const float*)d_in[2];   // (1,64,128,160)
  const float* pw_w0    = (const float*)d_in[4];
  const float* pw_g0    = (const float*)d_in[5];
  const float* pw_b0    = (const float*)d_in[6];
  const float* pw_w1    = (const float*)d_in[7];
  const float* pw_g1    = (const float*)d_in[8];
  const float* pw_b1    = (const float*)d_in[9];
  const float* pw_w2    = (const float*)d_in[10];
  const float* pw_bias2 = (const float*)d_in[11];
  const float* cr_w0    = (const float*)d_in[12];
  const float* cr_b0    = (const float*)d_in[13];
  const float* cr_w1    = (const float*)d_in[14];
  const float* cr_b1    = (const float*)d_in[15];

  float* out         = (float*)d_out;
  float* o_depth     = out;                       // (1,128,160)
  float* o_simdepth  = out + HWn;                 // (1,128,160)
  float* o_conf      = out + 2*HWn;               // (1,128,160)
  float* o_prob      = out + 3*HWn;               // (1,64,128,160)
  float* o_vw        = out + 3*HWn + NVOX;        // (1,4,128,160)

  float* ws      = (float*)d_ws;
  float* rt      = ws;                 // 48 floats (pad to 64)
  float* refT    = ws + 64;            // CHWn
  float* srcT    = refT + CHWn;        // CHWn
  float* simTmp  = srcT + CHWn;        // NVOX
  float* sigTmp  = simTmp + NVOX;      // NVOX
  float* simSum  = sigTmp + NVOX;      // NVOX
  float* wSum    = simSum + NVOX;      // HWn

  k_setup<<<1, 32, 0, stream>>>(projm, rt);
  k_transpose<<<CHWn/256, 256, 0, stream>>>(features, refT);
  k_init<<<NVOX/256, 256, 0, stream>>>(simSum, wSum);
  for (int v = 1; v < 5; ++v) {
    k_transpose<<<CHWn/256, 256, 0, stream>>>(features + (size_t)v*CHWn, srcT);
    k_warp_sim<<<NVOX/256, 256, 0, stream>>>(srcT, refT, depths, rt + (v-1)*12, simTmp);
    // 16 voxels/wave/step * 8 steps * 8 waves/block = 1024 voxels per block
    k_mlp<<<NVOX/1024, 256, 0, stream>>>(simTmp, pw_w0, pw_g0, pw_b0, pw_w1, pw_g1,
                                         pw_b1, pw_w2, pw_bias2, sigTmp);
    k_vw_accum<<<HWn/256, 256, 0, stream>>>(simTmp, sigTmp, simSum, wSum,
                                            o_vw + (size_t)(v-1)*HWn);
  }
  k_final<<<HWn/256, 256, 0, stream>>>(simSum, wSum, depths, cr_w0, cr_b0, cr_w1, cr_b1,
                                       o_depth, o_simdepth, o_conf, o_prob);
}